// My_Conv2D_9749575762014
// MI455X (gfx1250) — compile-verified
//
#include <hip/hip_runtime.h>

typedef __attribute__((ext_vector_type(2))) float v2f;
typedef __attribute__((ext_vector_type(8))) float v8f;

#define B_      32
#define H_      224
#define W_      224
#define CIN_    16
#define COUT_   32
#define SEGW    112           // pixels per block segment (2 segments per row)
#define LCOLS   (SEGW + 2)    // 114 columns incl. halo
#define CSTR    17            // padded channel stride (bank-conflict-free, 17 coprime w/ 64)
#define NTHREADS 224          // 7 wave32s, one 16-pixel tile each
#define NWT     (2 * 9 * 4 * 2 * 32)   // 4608 rearranged weights

__global__ __launch_bounds__(NTHREADS)
void conv3x3_wmma_f32_gfx1250(const float* __restrict__ in,
                              const float* __restrict__ wt,
                              float* __restrict__ out) {
    __shared__ float lds_in[3 * LCOLS * CSTR];   // 5814 floats = 23.3 KB
    __shared__ float lds_wt[NWT];                // 4608 floats = 18.4 KB

    const int tid = threadIdx.x;
    const int bid = blockIdx.x;
    const int seg = bid & 1;
    const int h   = (bid >> 1) % H_;
    const int b   = bid / (2 * H_);
    const int w0  = seg * SEGW;

    // ---- Stage weights into LDS, pre-swizzled to the WMMA B-operand lane layout.
    // flat = (((ch*9 + tap)*4 + kc)*2 + v)*32 + lane
    // value = wt[(tap*CIN + (kc*4 + (lane>>4)*2 + v)) * COUT + ch*16 + (lane&15)]
    for (int idx = tid; idx < NWT; idx += NTHREADS) {
        const int lane = idx & 31;
        const int v    = (idx >> 5) & 1;
        const int kc   = (idx >> 6) & 3;
        const int tap  = (idx >> 8) % 9;
        const int ch   = idx / 2304;
        const int k    = kc * 4 + ((lane >> 4) << 1) + v;
        const int n    = ch * 16 + (lane & 15);
        lds_wt[idx] = wt[(tap * CIN_ + k) * COUT_ + n];
    }

    // ---- Stage input halo tile: rows h-1..h+1, cols w0-1..w0+112, zero-padded (SAME).
    for (int idx = tid; idx < 3 * LCOLS; idx += NTHREADS) {
        const int r  = idx / LCOLS;
        const int c  = idx % LCOLS;
        const int gh = h + r - 1;
        const int gw = w0 + c - 1;
        float* dst = &lds_in[idx * CSTR];
        if (gh >= 0 && gh < H_ && gw >= 0 && gw < W_) {
            const float4* src =
                (const float4*)&in[(((long)b * H_ + gh) * W_ + gw) * CIN_];
            #pragma unroll
            for (int q = 0; q < 4; ++q) {
                const float4 vv = src[q];
                dst[q * 4 + 0] = vv.x;
                dst[q * 4 + 1] = vv.y;
                dst[q * 4 + 2] = vv.z;
                dst[q * 4 + 3] = vv.w;
            }
        } else {
            #pragma unroll
            for (int q = 0; q < 16; ++q) dst[q] = 0.0f;
        }
    }
    __syncthreads();

    // ---- Implicit GEMM: each wave computes 16 pixels x 32 couts.
    const int lane = tid & 31;
    const int wave = tid >> 5;          // 0..6
    const int m    = lane & 15;         // pixel within tile / N column
    const int kh2  = (lane >> 4) << 1;  // K sub-offset: 0 (lanes 0-15), 2 (lanes 16-31)

    v8f c0 = {};   // couts 0..15
    v8f c1 = {};   // couts 16..31

    #pragma unroll
    for (int tap = 0; tap < 9; ++tap) {
        const int kh = tap / 3;
        const int kw = tap % 3;
        const float* arow = &lds_in[(kh * LCOLS + wave * 16 + m + kw) * CSTR];
        const float* brow = &lds_wt[tap * 256];
        #pragma unroll
        for (int kc = 0; kc < 4; ++kc) {
            v2f a;
            a.x = arow[kc * 4 + kh2 + 0];   // A: 16x4 fp32, K split across lane halves
            a.y = arow[kc * 4 + kh2 + 1];
            v2f b0, b1;
            b0.x = brow[kc * 64 +      lane];
            b0.y = brow[kc * 64 + 32 + lane];
            b1.x = brow[2304 + kc * 64 +      lane];
            b1.y = brow[2304 + kc * 64 + 32 + lane];
            c0 = __builtin_amdgcn_wmma_f32_16x16x4_f32(
                     false, a, false, b0, (short)0, c0, false, false);
            c1 = __builtin_amdgcn_wmma_f32_16x16x4_f32(
                     false, a, false, b1, (short)0, c1, false, false);
        }
    }

    // ---- Store D: lane L, VGPR v -> row m = v + 8*(L>>4), col n = L&15.
    const int mrow = (lane >> 4) << 3;  // 0 or 8
    const int n    = lane & 15;
    float* obase = &out[(((long)b * H_ + h) * W_ + (w0 + wave * 16)) * COUT_];
    #pragma unroll
    for (int v = 0; v < 8; ++v) {
        obase[(mrow + v) * COUT_ + n]      = c0[v];
        obase[(mrow + v) * COUT_ + 16 + n] = c1[v];
    }
}

extern "C" void kernel_launch(void* const* d_in, const int* in_sizes, int n_in,
                              void* d_out, int out_size, void* d_ws, size_t ws_size,
                              hipStream_t stream) {
    (void)in_sizes; (void)n_in; (void)out_size; (void)d_ws; (void)ws_size;
    const float* in  = (const float*)d_in[0];   // (32,224,224,16) NHWC fp32
    const float* wt  = (const float*)d_in[1];   // (3,3,16,32) HWIO fp32
    float*       out = (float*)d_out;           // (32,224,224,32) NHWC fp32

    const int grid = B_ * H_ * 2;               // 14336 blocks
    conv3x3_wmma_f32_gfx1250<<<dim3(grid), dim3(NTHREADS), 0, stream>>>(in, wt, out);
}